// Renderer_72404558676846
// MI455X (gfx1250) — compile-verified
//
#include <hip/hip_runtime.h>
#include <hip/hip_bf16.h>
#include <math.h>
#include <stdint.h>

// NeRF ray-march (CDNA5 / gfx1250, wave32).
// Latency-bound scattered-gather workload: 16MB density grid lives in L2
// (192MB), so we (a) warm L2 with global_prefetch_b8, (b) stage the AoS ray
// data block-wise into LDS via global_load_async_to_lds_b128 (ASYNCcnt path),
// (c) read it back with explicit ds_load_b32 asm (compiler cannot see the
// async writes, so the reads must be opaque too), (d) march with branch-free
// per-step math and wave32 early termination.

#define GRID_N    128
#define MAX_STEPSI 128

__device__ __forceinline__ float clampf(float v, float lo, float hi) {
    return fminf(fmaxf(v, lo), hi);
}

__device__ __forceinline__ float safe_dir(float d) {
    // matches jnp.where(|d| > 1e-8, d, 1e-8)
    return (fabsf(d) > 1e-8f) ? d : 1e-8f;
}

__global__ __launch_bounds__(256) void nerf_march_kernel(
    const float* __restrict__ raysO,
    const float* __restrict__ raysD,
    const float* __restrict__ densityGrid,
    float* __restrict__ out,
    int nRays, int nPrefetchLines)
{
    __shared__ float sO[256 * 3];   // 3 KB
    __shared__ float sD[256 * 3];   // 3 KB

    const int tid = threadIdx.x;
    const int ray = blockIdx.x * 256 + tid;

    // ---------- CDNA5 async staging: 256 rays * 3 floats = 3072 B per array,
    // copied as 192 lanes x b128 (coalesced) into LDS. -----------------------
    const bool fullBlock = (blockIdx.x * 256 + 256) <= nRays;
    if (fullBlock) {
        const float* gO = raysO + (size_t)blockIdx.x * 768;
        const float* gD = raysD + (size_t)blockIdx.x * 768;
        if (tid < 192) {
            unsigned voff = (unsigned)tid * 16u;
            unsigned ldsO = (unsigned)(uintptr_t)(&sO[0]) + voff; // low 32 bits = LDS offset
            unsigned ldsD = (unsigned)(uintptr_t)(&sD[0]) + voff;
            asm volatile("global_load_async_to_lds_b128 %0, %1, %2"
                         :: "v"(ldsO), "v"(voff), "s"(gO) : "memory");
            asm volatile("global_load_async_to_lds_b128 %0, %1, %2"
                         :: "v"(ldsD), "v"(voff), "s"(gD) : "memory");
        }
        asm volatile("s_wait_asynccnt 0x0" ::: "memory");
    }

    // ---------- Warm L2 with the full 16 MB density grid: one 128B line per
    // thread (131072 threads * 128 B == 16 MB exactly). global_prefetch_b8
    // does not touch LOADcnt; it just pulls lines toward GL2. ----------------
    if (ray < nPrefetchLines) {
        __builtin_prefetch((const char*)densityGrid + (size_t)ray * 128u, 0, 1);
    }

    __syncthreads();

    if (ray >= nRays) return;

    float ox, oy, oz, dx, dy, dz;
    if (fullBlock) {
        // Explicit DS loads of the async-staged data, with the dscnt wait
        // fused into the same asm block so nothing can be scheduled between
        // the loads and the wait. Addresses are 4B-aligned (tid*12 + 0/4/8).
        unsigned aO = (unsigned)(uintptr_t)(&sO[0]) + (unsigned)tid * 12u;
        unsigned aD = (unsigned)(uintptr_t)(&sD[0]) + (unsigned)tid * 12u;
        asm volatile(
            "ds_load_b32 %0, %6\n\t"
            "ds_load_b32 %1, %6 offset:4\n\t"
            "ds_load_b32 %2, %6 offset:8\n\t"
            "ds_load_b32 %3, %7\n\t"
            "ds_load_b32 %4, %7 offset:4\n\t"
            "ds_load_b32 %5, %7 offset:8\n\t"
            "s_wait_dscnt 0x0"
            : "=&v"(ox), "=&v"(oy), "=&v"(oz),
              "=&v"(dx), "=&v"(dy), "=&v"(dz)
            : "v"(aO), "v"(aD)
            : "memory");
    } else {
        ox = raysO[ray * 3 + 0]; oy = raysO[ray * 3 + 1]; oz = raysO[ray * 3 + 2];
        dx = raysD[ray * 3 + 0]; dy = raysD[ray * 3 + 1]; dz = raysD[ray * 3 + 2];
    }

    // ---------- near/far from AABB [-2,2]^3 (slab method) -------------------
    const float ivx = 1.0f / safe_dir(dx);
    const float ivy = 1.0f / safe_dir(dy);
    const float ivz = 1.0f / safe_dir(dz);
    float t0 = (-2.0f - ox) * ivx, t1 = (2.0f - ox) * ivx;
    float tminx = fminf(t0, t1), tmaxx = fmaxf(t0, t1);
    t0 = (-2.0f - oy) * ivy; t1 = (2.0f - oy) * ivy;
    float tminy = fminf(t0, t1), tmaxy = fmaxf(t0, t1);
    t0 = (-2.0f - oz) * ivz; t1 = (2.0f - oz) * ivz;
    float tminz = fminf(t0, t1), tmaxz = fmaxf(t0, t1);
    const float tmin  = fmaxf(tminx, fmaxf(tminy, tminz));
    const float tmax  = fminf(tmaxx, fminf(tmaxy, tmaxz));
    const float nearT = fmaxf(tmin, 0.05f);
    const float farT  = fmaxf(tmax, nearT + 1e-6f);
    const float dt    = (farT - nearT) * (1.0f / 128.0f);

    const float* __restrict__ g0 = densityGrid;                                    // cascade 0
    const float* __restrict__ g1 = densityGrid + (size_t)GRID_N * GRID_N * GRID_N; // cascade 1

    float T = 1.0f, accR = 0.0f, accG = 0.0f, accB = 0.0f, wsum = 0.0f;

    #pragma unroll 4
    for (int s = 0; s < MAX_STEPSI; ++s) {
        const float t = fmaf((float)s + 0.5f, dt, nearT);
        const float x = fmaf(t, dx, ox);
        const float y = fmaf(t, dy, oy);
        const float z = fmaf(t, dz, oz);

        // cascade: ceil(log2(max(|xyz|,1))) clipped to [0,1]  ==  (mx > 1)
        const float mx   = fmaxf(fmaxf(fabsf(x), fabsf(y)), fabsf(z));
        const bool  cas1 = (mx > 1.0f);
        const float scl  = cas1 ? 0.25f : 0.5f;          // 0.5 / bound_c

        const float hi = 1.0f - 1e-6f;
        const int ix = (int)(clampf(fmaf(x, scl, 0.5f), 0.0f, hi) * 128.0f);
        const int iy = (int)(clampf(fmaf(y, scl, 0.5f), 0.0f, hi) * 128.0f);
        const int iz = (int)(clampf(fmaf(z, scl, 0.5f), 0.0f, hi) * 128.0f);

        const float* __restrict__ gp = cas1 ? g1 : g0;
        const float occv = gp[(size_t)(((ix * GRID_N) + iy) * GRID_N + iz)];   // L2-resident gather

        const float d2    = fmaf(x, x, fmaf(y, y, z * z));
        const float blob  = 5.0f * __expf(-12.5f * d2);   // 1/(2*0.2^2) = 12.5
        const float sigma = (occv > 10.0f) ? blob : 0.0f;

        const float alpha = 1.0f - __expf(-sigma * dt);
        const float w     = alpha * T;

        accR = fmaf(w, clampf(fmaf(x, 0.25f, 0.5f), 0.0f, 1.0f), accR);
        accG = fmaf(w, clampf(fmaf(y, 0.25f, 0.5f), 0.0f, 1.0f), accG);
        accB = fmaf(w, clampf(fmaf(z, 0.25f, 0.5f), 0.0f, 1.0f), accB);
        wsum += w;
        T *= (1.0f - alpha);

        // wave32 early exit; residual weight <= T < 1e-4 (telescoping sum)
        if ((s & 15) == 15 && __all(T < 1e-4f)) break;
    }

    // background composite with bgColor == 1
    const float bg = 1.0f - wsum;
    out[ray * 3 + 0] = accR + bg;
    out[ray * 3 + 1] = accG + bg;
    out[ray * 3 + 2] = accB + bg;
}

extern "C" void kernel_launch(void* const* d_in, const int* in_sizes, int n_in,
                              void* d_out, int out_size, void* d_ws, size_t ws_size,
                              hipStream_t stream) {
    const float* raysO = (const float*)d_in[0];   // [N,3] f32
    const float* raysD = (const float*)d_in[1];   // [N,3] f32
    const float* grid  = (const float*)d_in[2];   // [2,128,128,128] f32
    float* out = (float*)d_out;                   // [N,3] f32

    const int nRays  = in_sizes[0] / 3;
    const int nLines = (int)(((long long)in_sizes[2] * (long long)sizeof(float)) / 128);
    const int blocks = (nRays + 255) / 256;

    hipLaunchKernelGGL(nerf_march_kernel, dim3(blocks), dim3(256), 0, stream,
                       raysO, raysD, grid, out, nRays, nLines);
}